// GNN_24850680775342
// MI455X (gfx1250) — compile-verified
//
#include <hip/hip_runtime.h>

// ---------------------------------------------------------------------------
// 2-layer GCN (PyG GCNConv, eval) + linear head for MI455X (gfx1250, wave32).
//  - Dense 128x128 transforms use V_WMMA_F32_16X16X4_F32 (fp32-exact).
//  - Edge aggregation is the memory-bound part: coalesced gather + f32 atomics.
// ---------------------------------------------------------------------------

typedef __attribute__((ext_vector_type(2))) float v2f;
typedef __attribute__((ext_vector_type(8))) float v8f;

#define NF 128   // feature width of both GCN layers
#define NCLS 10

// ---------------- degree / normalization ----------------

__global__ void deg_init_kernel(float* __restrict__ deg, int n) {
  int i = blockIdx.x * blockDim.x + threadIdx.x;
  if (i < n) deg[i] = 1.0f;  // self-loop weight 1
}

__global__ void deg_accum_kernel(const long long* __restrict__ dst,
                                 const float* __restrict__ w,
                                 float* __restrict__ deg, int e) {
  int i = blockIdx.x * blockDim.x + threadIdx.x;
  if (i < e) atomicAdd(&deg[(int)dst[i]], w[i]);
}

__global__ void deg_rsqrt_kernel(float* __restrict__ deg, int n) {
  int i = blockIdx.x * blockDim.x + threadIdx.x;
  if (i < n) {
    float d = deg[i];
    deg[i] = (d > 0.0f) ? rsqrtf(d) : 0.0f;   // becomes dinv
  }
}

__global__ void norm_kernel(const long long* __restrict__ src,
                            const long long* __restrict__ dst,
                            const float* __restrict__ w,
                            const float* __restrict__ dinv,
                            float* __restrict__ norm, int e) {
  int i = blockIdx.x * blockDim.x + threadIdx.x;
  if (i < e) norm[i] = dinv[(int)src[i]] * w[i] * dinv[(int)dst[i]];
}

// ---------------- dense GEMM: H[n,128] = X[n,128] @ W[128,128] ----------------
// One block = 256 threads = 8 waves; block computes rows [r0, r0+16) x 128 cols.
// Wave w computes the 16x16 tile at columns [16w, 16w+16) with 32 chained
// v_wmma_f32_16x16x4_f32 ops (K = 128).
__global__ void __launch_bounds__(256)
gemm128_wmma_kernel(const float* __restrict__ X, const float* __restrict__ W,
                    float* __restrict__ H, int nrows) {
  __shared__ float sX[16 * NF];       // 8 KB  A-tile
  __shared__ float sW[32 * NF];       // 16 KB B-chunk (32 K-rows)

  const int r0 = blockIdx.x * 16;
  const int t  = threadIdx.x;
  const int rowsHere = nrows - r0 < 16 ? nrows - r0 : 16;

  // Stage X tile (16x128) with coalesced float4 loads.
  for (int i = t; i < 16 * NF / 4; i += 256) {
    int r  = i >> 5;          // 32 float4 per row
    int c4 = i & 31;
    float4 v = make_float4(0.f, 0.f, 0.f, 0.f);
    if (r < rowsHere) v = ((const float4*)(X + (size_t)(r0 + r) * NF))[c4];
    ((float4*)sX)[i] = v;
  }

  const int lane = t & 31;
  const int wave = t >> 5;            // 0..7 -> column tile
  const int colBase = wave * 16;
  const int half = lane >> 4;         // 0: lanes 0-15, 1: lanes 16-31
  const int l16  = lane & 15;

  v8f acc = {};

  for (int kc = 0; kc < NF; kc += 32) {
    __syncthreads();
    // Stage W[kc .. kc+32)[0..128) : 4096 floats, coalesced float4.
    const float4* Wc4 = (const float4*)(W + (size_t)kc * NF);
    for (int i = t; i < 32 * NF / 4; i += 256) ((float4*)sW)[i] = Wc4[i];
    if (kc + 32 < NF)  // hint next chunk into cache (global_prefetch_b8)
      __builtin_prefetch(W + (size_t)(kc + 32) * NF + t, 0, 3);
    __syncthreads();

    for (int k = 0; k < 32; k += 4) {
      // A frag (16x4 f32): lanes 0-15 hold {K=k,k+1} of row l16,
      //                    lanes 16-31 hold {K=k+2,k+3}.
      const float* ap = sX + l16 * NF + k + half * 2;
      v2f a; a.x = ap[0]; a.y = ap[1];
      // B frag (4x16 f32): v0 = row K (lanes 0-15) / K+2 (lanes 16-31),
      //                    v1 = row K+1 / K+3; column = colBase + l16.
      const float* bp = sW + (k + half * 2) * NF + colBase + l16;
      v2f b; b.x = bp[0]; b.y = bp[NF];
      acc = __builtin_amdgcn_wmma_f32_16x16x4_f32(
          /*neg_a=*/false, a, /*neg_b=*/false, b,
          /*c_mod=*/(short)0, acc, /*reuse_a=*/false, /*reuse_b=*/false);
    }
  }

  // C/D layout: VGPR v -> row v (lanes 0-15) and row v+8 (lanes 16-31), col l16.
  for (int v = 0; v < 8; ++v) {
    int m = v + half * 8;
    if (m < rowsHere)
      H[(size_t)(r0 + m) * NF + colBase + l16] = acc[v];
  }
}

// ---------------- aggregation ----------------

// out[i,f] = dinv[i]^2 * H[i,f]   (self-loop term, also initializes out)
__global__ void self_init_kernel(const float* __restrict__ H,
                                 const float* __restrict__ dinv,
                                 float* __restrict__ out, int n) {
  long long tid = (long long)blockIdx.x * blockDim.x + threadIdx.x;
  if (tid >= (long long)n * NF) return;
  int i = (int)(tid >> 7);
  float di = dinv[i];
  out[tid] = di * di * H[tid];
}

// out[dst,f] += norm[e] * H[src,f]; 128 consecutive threads per edge ->
// coalesced gather and coalesced global_atomic_add_f32 scatter.
__global__ void agg_edges_kernel(const float* __restrict__ H,
                                 const long long* __restrict__ src,
                                 const long long* __restrict__ dst,
                                 const float* __restrict__ norm,
                                 float* __restrict__ out, int e) {
  long long tid = (long long)blockIdx.x * blockDim.x + threadIdx.x;
  long long ei = tid >> 7;
  int f = (int)(tid & (NF - 1));
  if (ei >= e) return;
  int s = (int)src[ei];
  int d = (int)dst[ei];
  float v = norm[ei] * H[(size_t)s * NF + f];
  atomicAdd(out + (size_t)d * NF + f, v);
}

// y[i,f] = relu(x[i,f] + b[f])
__global__ void bias_relu_kernel(const float* __restrict__ x,
                                 const float* __restrict__ b,
                                 float* __restrict__ y, int n) {
  long long tid = (long long)blockIdx.x * blockDim.x + threadIdx.x;
  if (tid >= (long long)n * NF) return;
  float v = x[tid] + b[(int)(tid & (NF - 1))];
  y[tid] = v > 0.0f ? v : 0.0f;
}

// ---------------- classifier: out[n,10] = H[n,128] @ Wl[128,10] + bl ----------------
__global__ void __launch_bounds__(256)
classifier_kernel(const float* __restrict__ H, const float* __restrict__ Wl,
                  const float* __restrict__ bl, float* __restrict__ out, int n) {
  __shared__ float sW[NF * NCLS];
  __shared__ float sb[NCLS];
  int t = threadIdx.x;
  for (int i = t; i < NF * NCLS; i += 256) sW[i] = Wl[i];
  if (t < NCLS) sb[t] = bl[t];
  __syncthreads();
  int node = blockIdx.x * 256 + t;
  if (node >= n) return;
  const float* h = H + (size_t)node * NF;
  float acc[NCLS];
#pragma unroll
  for (int c = 0; c < NCLS; ++c) acc[c] = sb[c];
  for (int k = 0; k < NF; ++k) {
    float hv = h[k];
#pragma unroll
    for (int c = 0; c < NCLS; ++c) acc[c] += hv * sW[k * NCLS + c];
  }
#pragma unroll
  for (int c = 0; c < NCLS; ++c) out[(size_t)node * NCLS + c] = acc[c];
}

// ---------------- host-side pipeline ----------------

extern "C" void kernel_launch(void* const* d_in, const int* in_sizes, int n_in,
                              void* d_out, int out_size, void* d_ws, size_t ws_size,
                              hipStream_t stream) {
  const float*     x   = (const float*)d_in[0];
  const long long* ei  = (const long long*)d_in[1];   // int64 [2, E]
  const float*     w   = (const float*)d_in[2];
  const float*     W1  = (const float*)d_in[3];
  const float*     b1  = (const float*)d_in[4];
  const float*     W2  = (const float*)d_in[5];
  const float*     b2  = (const float*)d_in[6];
  const float*     Wl  = (const float*)d_in[7];
  const float*     bl  = (const float*)d_in[8];
  float* out = (float*)d_out;

  const int n = in_sizes[0] / NF;     // 100000
  const int e = in_sizes[2];          // 1600000
  const long long* srcI = ei;         // row 0
  const long long* dstI = ei + e;     // row 1

  // workspace layout (floats): dinv[n] | norm[e] | bufA[n*128] | bufB[n*128]
  float* dinv = (float*)d_ws;
  float* norm = dinv + n;
  float* bufA = norm + e;             // offset (n+e)*4 B, 16B-aligned for n,e here
  float* bufB = bufA + (size_t)n * NF;

  const int T = 256;
  const int gN   = (n + T - 1) / T;
  const int gE   = (e + T - 1) / T;
  const int gNF  = (int)(((long long)n * NF + T - 1) / T);
  const int gENF = (int)(((long long)e * NF + T - 1) / T);
  const int gGemm = (n + 15) / 16;

  // normalization coefficients (shared by both layers)
  deg_init_kernel<<<gN, T, 0, stream>>>(dinv, n);
  deg_accum_kernel<<<gE, T, 0, stream>>>(dstI, w, dinv, e);
  deg_rsqrt_kernel<<<gN, T, 0, stream>>>(dinv, n);
  norm_kernel<<<gE, T, 0, stream>>>(srcI, dstI, w, dinv, norm, e);

  // ----- layer 1: H1 = x @ W1 ; agg ; +b1 ; relu -> bufA -----
  gemm128_wmma_kernel<<<gGemm, T, 0, stream>>>(x, W1, bufA, n);
  self_init_kernel<<<gNF, T, 0, stream>>>(bufA, dinv, bufB, n);
  agg_edges_kernel<<<gENF, T, 0, stream>>>(bufA, srcI, dstI, norm, bufB, e);
  bias_relu_kernel<<<gNF, T, 0, stream>>>(bufB, b1, bufA, n);

  // ----- layer 2: H2 = bufA @ W2 ; agg ; +b2 ; relu -> bufA -----
  gemm128_wmma_kernel<<<gGemm, T, 0, stream>>>(bufA, W2, bufB, n);
  self_init_kernel<<<gNF, T, 0, stream>>>(bufB, dinv, bufA, n);
  agg_edges_kernel<<<gENF, T, 0, stream>>>(bufB, srcI, dstI, norm, bufA, e);
  bias_relu_kernel<<<gNF, T, 0, stream>>>(bufA, b2, bufA, n);

  // ----- head -----
  classifier_kernel<<<gN, T, 0, stream>>>(bufA, Wl, bl, out, n);
}